// AgeUGP_v2_18081812317002
// MI455X (gfx1250) — compile-verified
//
#include <hip/hip_runtime.h>

#define N_SNPS   200000
#define N_GENES  20000
#define N_NODES  1000000
#define N_FILTERS 8
#define D_HIDDEN 64
#define FEAT_DIM 16
#define MAIN_DIM 15
#define BATCH    16
#define BN_EPS   1e-5f

typedef __attribute__((ext_vector_type(2))) float v2f;
typedef __attribute__((ext_vector_type(8))) float v8f;

// ---------------------------------------------------------------------------
// Stage 1: coeff[s] = mean over 8 filters of filters[f][s]
// ---------------------------------------------------------------------------
__global__ void k_coeff(const float* __restrict__ filters, float* __restrict__ coeff) {
    int s = blockIdx.x * blockDim.x + threadIdx.x;
    if (s < N_SNPS) {
        float c = 0.f;
#pragma unroll
        for (int f = 0; f < N_FILTERS; ++f) c += filters[f * N_SNPS + s];
        coeff[s] = c * (1.0f / N_FILTERS);
    }
}

__global__ void k_zero(float* __restrict__ p, int n) {
    int i = blockIdx.x * blockDim.x + threadIdx.x;
    if (i < n) p[i] = 0.f;
}

__global__ void k_init_pre1(const float* __restrict__ b1, float* __restrict__ pre1) {
    for (int i = threadIdx.x; i < BATCH * D_HIDDEN; i += blockDim.x)
        pre1[i] = b1[i & (D_HIDDEN - 1)];
}

// ---------------------------------------------------------------------------
// Stage 2: sorted segment-sum gather.
// sample_h[b,g] = sum_{n in gene g} snp[b, sid[n]] * coeff[sid[n]]
// node2gene is sorted, so each thread walks a contiguous node chunk keeping a
// register accumulator, flushing with a HW f32 global atomic only on gene
// boundaries (~50 nodes/gene => atomics are rare).
// Lanes 0..15 of a wave share a chunk (b = lane id) so id loads broadcast.
// ---------------------------------------------------------------------------
#define NPT 128
#define NCHUNK ((N_NODES + NPT - 1) / NPT)          // 7813
#define SCATTER_THREADS (BATCH * NCHUNK)            // 125008

__global__ void k_scatter(const float* __restrict__ snp,
                          const float* __restrict__ coeff,
                          const int* __restrict__ snp_ids,
                          const int* __restrict__ node2gene,
                          float* __restrict__ sample_h) {
    int tid = blockIdx.x * blockDim.x + threadIdx.x;
    if (tid >= SCATTER_THREADS) return;
    int b     = tid & (BATCH - 1);
    int chunk = tid >> 4;
    int n0 = chunk * NPT;
    int n1 = min(n0 + NPT, N_NODES);
    const float* __restrict__ srow = snp + (size_t)b * N_SNPS;
    float* __restrict__ shrow = sample_h + (size_t)b * N_GENES;

    int   curg = node2gene[n0];
    float acc  = 0.f;
    for (int n = n0; n < n1; ++n) {
        int g = node2gene[n];
        if (g != curg) {
            unsafeAtomicAdd(&shrow[curg], acc);   // global_atomic_add_f32
            acc = 0.f;
            curg = g;
        }
        int sid = snp_ids[n];
        acc = fmaf(srow[sid], coeff[sid], acc);
    }
    unsafeAtomicAdd(&shrow[curg], acc);
}

// ---------------------------------------------------------------------------
// Stage 3: GEMM1 pre1 = sample_h(16x20000) @ W1.T(20000x64)  [+ b1 pre-init]
// Full-f32 WMMA 16x16x4. One workgroup = 4 waves = the 4 N-tiles of 64.
// K split across KBLK blocks; partial 16x16 tiles reduced with f32 atomics.
// f32 16x4 A fragment  : lane(m,half) holds A[m][2h], A[m][2h+1]
// f32 4x16 B fragment  : lane(n,half) holds B[2h][n], B[2h+1][n] = W1[n][2h..]
// f32 16x16 D fragment : VGPR v, lane(n,half) -> D[v+8h][n]
// ---------------------------------------------------------------------------
#define KBLK 40
#define KSLICE (N_GENES / KBLK)   // 500, multiple of 4

__global__ void k_gemm1(const float* __restrict__ sample_h,
                        const float* __restrict__ W1,
                        float* __restrict__ pre1) {
    int lane = threadIdx.x & 31;
    int wave = threadIdx.x >> 5;          // 0..3 -> N tile
    int mn   = lane & 15;                 // A row (batch) == B col index
    int half = lane >> 4;
    int ncol = wave * 16 + mn;            // hidden unit 0..63

    const float* __restrict__ arow = sample_h + (size_t)mn * N_GENES + 2 * half;
    const float* __restrict__ brow = W1 + (size_t)ncol * N_GENES + 2 * half;

    int k0 = blockIdx.x * KSLICE;
    v8f c = {};
    for (int kb = k0; kb < k0 + KSLICE; kb += 4) {
        v2f a = *(const v2f*)(arow + kb);
        v2f bm = *(const v2f*)(brow + kb);
        c = __builtin_amdgcn_wmma_f32_16x16x4_f32(
                /*neg_a=*/false, a, /*neg_b=*/false, bm,
                /*c_mod=*/(short)0, c, /*reuse_a=*/false, /*reuse_b=*/false);
    }
#pragma unroll
    for (int v = 0; v < 8; ++v) {
        int row = v + 8 * half;           // batch index
        unsafeAtomicAdd(&pre1[row * D_HIDDEN + ncol], c[v]);
    }
}

// ---------------------------------------------------------------------------
// Stage 4 (single block): BN1+relu -> WMMA GEMM2 (16x64x16) -> BN2+relu ->
// slice[:15] @ W3.T + b3 -> logits(16)
// ---------------------------------------------------------------------------
__global__ void k_tail(const float* __restrict__ pre1,
                       const float* __restrict__ g1, const float* __restrict__ bb1,
                       const float* __restrict__ W2, const float* __restrict__ b2,
                       const float* __restrict__ g2, const float* __restrict__ bb2,
                       const float* __restrict__ W3, const float* __restrict__ b3,
                       float* __restrict__ out) {
    __shared__ float hbuf[BATCH][D_HIDDEN];
    __shared__ float featbuf[BATCH][FEAT_DIM];
    int t = threadIdx.x;

    // BN1 over batch axis per hidden column, then relu
    if (t < D_HIDDEN) {
        float sum = 0.f, sum2 = 0.f;
#pragma unroll
        for (int b = 0; b < BATCH; ++b) {
            float x = pre1[b * D_HIDDEN + t];
            sum += x; sum2 += x * x;
        }
        float mu  = sum  * (1.0f / BATCH);
        float var = sum2 * (1.0f / BATCH) - mu * mu;
        float scale = g1[t] * rsqrtf(var + BN_EPS);
        float beta  = bb1[t];
#pragma unroll
        for (int b = 0; b < BATCH; ++b) {
            float y = (pre1[b * D_HIDDEN + t] - mu) * scale + beta;
            hbuf[b][t] = y > 0.f ? y : 0.f;
        }
    }
    __syncthreads();

    // GEMM2 on wave 0 (threads 0..31 => EXEC all ones for that wave):
    // featbuf_pre = hbuf(16x64) @ W2.T(64x16) + b2
    if (t < 32) {
        int mn = t & 15;        // batch row == output col index
        int half = t >> 4;
        v8f c;
#pragma unroll
        for (int v = 0; v < 8; ++v) c[v] = b2[mn];   // all comps share N=mn
#pragma unroll
        for (int kb = 0; kb < D_HIDDEN; kb += 4) {
            v2f a, bm;
            a.x  = hbuf[mn][kb + 2 * half];
            a.y  = hbuf[mn][kb + 2 * half + 1];
            bm.x = W2[mn * D_HIDDEN + kb + 2 * half];      // B[k][n] = W2[n][k]
            bm.y = W2[mn * D_HIDDEN + kb + 2 * half + 1];
            c = __builtin_amdgcn_wmma_f32_16x16x4_f32(
                    false, a, false, bm, (short)0, c, false, false);
        }
#pragma unroll
        for (int v = 0; v < 8; ++v) featbuf[v + 8 * half][mn] = c[v];
    }
    __syncthreads();

    // BN2 over batch axis per feature column, then relu (in place)
    if (t < FEAT_DIM) {
        float sum = 0.f, sum2 = 0.f;
#pragma unroll
        for (int b = 0; b < BATCH; ++b) {
            float x = featbuf[b][t];
            sum += x; sum2 += x * x;
        }
        float mu  = sum  * (1.0f / BATCH);
        float var = sum2 * (1.0f / BATCH) - mu * mu;
        float scale = g2[t] * rsqrtf(var + BN_EPS);
        float beta  = bb2[t];
#pragma unroll
        for (int b = 0; b < BATCH; ++b) {
            float y = (featbuf[b][t] - mu) * scale + beta;
            featbuf[b][t] = y > 0.f ? y : 0.f;
        }
    }
    __syncthreads();

    // logits[b] = feat[b,:15] . W3 + b3
    if (t < BATCH) {
        float acc = b3[0];
#pragma unroll
        for (int j = 0; j < MAIN_DIM; ++j) acc = fmaf(featbuf[t][j], W3[j], acc);
        out[t] = acc;
    }
}

// ---------------------------------------------------------------------------
// Host launcher
// inputs: 0 snp, 1 filters, 2 W1, 3 b1, 4 g1, 5 bb1, 6 W2, 7 b2, 8 g2,
//         9 bb2, 10 W3, 11 b3, 12 snp_ids(int), 13 node2gene(int)
// ws layout (floats): [0,200000) coeff | [200000,520000) sample_h |
//                     [520000,521024) pre1     (~2.08 MB total)
// ---------------------------------------------------------------------------
extern "C" void kernel_launch(void* const* d_in, const int* in_sizes, int n_in,
                              void* d_out, int out_size, void* d_ws, size_t ws_size,
                              hipStream_t stream) {
    const float* snp     = (const float*)d_in[0];
    const float* filters = (const float*)d_in[1];
    const float* W1      = (const float*)d_in[2];
    const float* b1      = (const float*)d_in[3];
    const float* g1      = (const float*)d_in[4];
    const float* bb1     = (const float*)d_in[5];
    const float* W2      = (const float*)d_in[6];
    const float* b2      = (const float*)d_in[7];
    const float* g2      = (const float*)d_in[8];
    const float* bb2     = (const float*)d_in[9];
    const float* W3      = (const float*)d_in[10];
    const float* b3      = (const float*)d_in[11];
    const int*   snp_ids   = (const int*)d_in[12];
    const int*   node2gene = (const int*)d_in[13];
    float* out = (float*)d_out;

    float* ws       = (float*)d_ws;
    float* coeff    = ws;                       // 200000
    float* sample_h = ws + N_SNPS;              // 320000
    float* pre1     = sample_h + BATCH * N_GENES; // 1024

    k_coeff<<<(N_SNPS + 255) / 256, 256, 0, stream>>>(filters, coeff);
    k_zero<<<(BATCH * N_GENES + 255) / 256, 256, 0, stream>>>(sample_h, BATCH * N_GENES);
    k_init_pre1<<<1, 256, 0, stream>>>(b1, pre1);
    k_scatter<<<(SCATTER_THREADS + 255) / 256, 256, 0, stream>>>(
        snp, coeff, snp_ids, node2gene, sample_h);
    k_gemm1<<<KBLK, 128, 0, stream>>>(sample_h, W1, pre1);
    k_tail<<<1, 256, 0, stream>>>(pre1, g1, bb1, W2, b2, g2, bb2, W3, b3, out);
}